// ContextMomentumNet_45999099740270
// MI455X (gfx1250) — compile-verified
//
#include <hip/hip_runtime.h>

typedef __attribute__((ext_vector_type(2))) float v2f;
typedef __attribute__((ext_vector_type(8))) float v8f;

// Cross-lane gather via the LDS permute network (no LDS memory touched).
__device__ __forceinline__ float lane_gather(float v, int byteIdx) {
  return __builtin_bit_cast(
      float, __builtin_amdgcn_ds_bpermute(byteIdx, __builtin_bit_cast(int, v)));
}

// One wave (32 lanes) per batch element b.
// metric_root[b]   = A^T G  with A[r=(l,j), i] = J[l,b,j,i]  (96 x 32)
//                    and   G[r=(l,j), m] = s_l * sum_k M[l,b,j,k] J[l,b,k,m]
// momentum_root[b] = sum_r A[r, i] * (s_l * p[r])
__global__ __launch_bounds__(256, 2) void pullback_wmma_kernel(
    const float* __restrict__ jac,   // [L,B,6,32]
    const float* __restrict__ mmt,   // [L,B,6]
    const float* __restrict__ met,   // [L,B,6,6]
    const float* __restrict__ scl,   // [L]
    float* __restrict__ out_mom,     // [B,32]
    float* __restrict__ out_met,     // [B,32,32]
    int B, int L) {
  constexpr int D = 6, C = 32;
  const int lane = threadIdx.x & 31;
  int b = (int)blockIdx.x * ((int)blockDim.x >> 5) + ((int)threadIdx.x >> 5);
  b = __builtin_amdgcn_readfirstlane(b);  // wave-uniform -> scalar loads for M/p/s
  if (b >= B) return;                      // uniform per wave; EXEC stays all-ones

  const int li = lane & 15;
  const int hi16 = lane >> 4;  // 0: lanes 0-15, 1: lanes 16-31
  const int idxt[2] = {li * 4, (16 + li) * 4};

  v8f acc[2][2] = {};   // 32x32 fp32 accumulator, four 16x16 WMMA tiles
  float momacc = 0.0f;  // momentum_root[b, lane]

  for (int lp = 0; lp < L; lp += 2) {
    // 12 rows (two leaves) of A and G, held as: register = row, lane = column.
    float Jr[12], Gr[12];
#pragma unroll
    for (int ll = 0; ll < 2; ++ll) {
      const int l = lp + ll;
      if (l >= L) {  // odd-L tail: pad with zero rows
#pragma unroll
        for (int k = 0; k < D; ++k) { Jr[D * ll + k] = 0.0f; Gr[D * ll + k] = 0.0f; }
        continue;
      }
      const float s = scl[l];
      const float* Jb = jac + ((size_t)l * B + b) * (D * C);
      const float* Mb = met + ((size_t)l * B + b) * (D * D);
      const float* pb = mmt + ((size_t)l * B + b) * D;
#pragma unroll
      for (int k = 0; k < D; ++k)  // coalesced 128B row loads
        Jr[D * ll + k] = Jb[k * C + lane];
#pragma unroll
      for (int j = 0; j < D; ++j) {
        float g = 0.0f;
#pragma unroll
        for (int k = 0; k < D; ++k)  // Mb[] is scalar (uniform) -> SMEM
          g = fmaf(Mb[j * D + k], Jr[D * ll + k], g);
        Gr[D * ll + j] = s * g;
        momacc = fmaf(s * pb[j], Jr[D * ll + j], momacc);
      }
    }

    // Three K-steps of 4 rows feed V_WMMA_F32_16X16X4_F32.
    // A 16x4 layout: lanes 0-15 -> {VGPR0:K0, VGPR1:K1}, lanes 16-31 -> {K2, K3}.
    // B 4x16 layout mirrors it (row striped across lanes within a VGPR).
#pragma unroll
    for (int ks = 0; ks < 3; ++ks) {
      const int rb = 4 * ks;
      v2f Aop[2], Bop[2];
#pragma unroll
      for (int t = 0; t < 2; ++t) {  // t = 16-wide tile index (ti for A, tm for B)
        const int ix = idxt[t];
        const float a0l = lane_gather(Jr[rb + 0], ix);
        const float a0h = lane_gather(Jr[rb + 2], ix);
        const float a1l = lane_gather(Jr[rb + 1], ix);
        const float a1h = lane_gather(Jr[rb + 3], ix);
        const float b0l = lane_gather(Gr[rb + 0], ix);
        const float b0h = lane_gather(Gr[rb + 2], ix);
        const float b1l = lane_gather(Gr[rb + 1], ix);
        const float b1h = lane_gather(Gr[rb + 3], ix);
        Aop[t].x = hi16 ? a0h : a0l;
        Aop[t].y = hi16 ? a1h : a1l;
        Bop[t].x = hi16 ? b0h : b0l;
        Bop[t].y = hi16 ? b1h : b1l;
      }
#pragma unroll
      for (int ti = 0; ti < 2; ++ti)
#pragma unroll
        for (int tm = 0; tm < 2; ++tm)
          acc[ti][tm] = __builtin_amdgcn_wmma_f32_16x16x4_f32(
              false, Aop[ti], false, Bop[tm], (short)0, acc[ti][tm], false, false);
    }
  }

  out_mom[(size_t)b * C + lane] = momacc;
  // C/D 16x16 layout: VGPR v -> row v (lanes 0-15) / v+8 (lanes 16-31), col = lane%16.
  float* om = out_met + (size_t)b * C * C;
#pragma unroll
  for (int ti = 0; ti < 2; ++ti)
#pragma unroll
    for (int tm = 0; tm < 2; ++tm)
#pragma unroll
      for (int v = 0; v < 8; ++v) {
        const int i = 16 * ti + v + 8 * hi16;
        const int m = 16 * tm + li;
        om[i * C + m] = acc[ti][tm][v];
      }
}

extern "C" void kernel_launch(void* const* d_in, const int* in_sizes, int n_in,
                              void* d_out, int out_size, void* d_ws, size_t ws_size,
                              hipStream_t stream) {
  // setup_inputs order: state[B,C], jacobians[L,B,6,32], momentums[L,B,6],
  //                     metrics[L,B,6,6], scaling[L]  (all fp32)
  const float* jac = (const float*)d_in[1];
  const float* mmt = (const float*)d_in[2];
  const float* met = (const float*)d_in[3];
  const float* scl = (const float*)d_in[4];
  const int C = 32;
  const int B = in_sizes[0] / C;  // state is [B, C]
  const int L = in_sizes[4];      // scaling is [L]
  float* out_mom = (float*)d_out;                 // [B, 32]
  float* out_met = out_mom + (size_t)B * C;       // [B, 32, 32]

  const int WPB = 8;  // 8 wave32s per block, one batch element per wave
  dim3 block(32 * WPB);
  dim3 grid((B + WPB - 1) / WPB);
  pullback_wmma_kernel<<<grid, block, 0, stream>>>(jac, mmt, met, scl, out_mom,
                                                   out_met, B, L);
}